// IFSFractal_30880814858732
// MI455X (gfx1250) — compile-verified
//
#include <hip/hip_runtime.h>

typedef __attribute__((ext_vector_type(2))) float v2f;
typedef __attribute__((ext_vector_type(4))) float v4f;
typedef __attribute__((ext_vector_type(8))) float v8f;

#define WAVES_PER_BLOCK 8
#define PTS_PER_WAVE    32
#define ROWSTRIDE       36   // 32 result rows + pad, multiple of 4 -> 16B-aligned columns

struct f2 { float x, y; };

// Branchless SELU: scale*max(x,0) + scale*alpha*(exp(min(x,0)) - 1)
// (exact: for x>0 the exp term is exp(0)-1 == 0; for x<=0 the max term is 0)
__device__ __forceinline__ float selu_f(float x) {
    const float scale       = 1.0507009873554804934f;
    const float alpha_scale = 1.7580993408473768599f;  // scale * alpha
    float pos = fmaxf(x, 0.0f);
    float neg = fminf(x, 0.0f);
    return scale * pos + alpha_scale * (__expf(neg) - 1.0f);
}

__device__ __forceinline__ void store8(float* dst, v8f d) {
    union { v8f v; v4f q[2]; } u;
    u.v = d;
    *(v4f*)(dst + 0) = u.q[0];
    *(v4f*)(dst + 4) = u.q[1];
}

__global__ __launch_bounds__(256) void IFSFractal_wmma_kernel(
    const float* __restrict__ points,       // [N,3]
    const float* __restrict__ prev_colors,  // [N,3]
    const int*   __restrict__ choices,      // [N]
    const float* __restrict__ matrices,     // [K,3,3]
    const float* __restrict__ biases,       // [K,3]
    const float* __restrict__ colors,       // [K,3]
    float*       __restrict__ out,          // tp [N,3] ++ new_colors [N,3]
    int N)
{
    __shared__ __align__(16) float lds[WAVES_PER_BLOCK * PTS_PER_WAVE * ROWSTRIDE];

    const int  lane = threadIdx.x & 31;
    const int  wib  = threadIdx.x >> 5;
    const bool hiL  = lane >= 16;        // lane half holds K=2,3 of A/B operands
    const int  m    = lane & 15;         // A row / B column

    // ---- Build A tiles once per wave: 16x4 f32, row m = (transform c, component e)
    // A[m][0..2] = M_c[k][e] (contract over k), A[m][3] = b_c[e]; rows 12..15 = 0
    v2f A0 = {0.0f, 0.0f};  // transforms 0..3
    v2f A1 = {0.0f, 0.0f};  // transforms 4..7
    if (m < 12) {
        const int c = m / 3;
        const int e = m - 3 * c;
        const float* M0 = matrices + c * 9;
        const float* M1 = matrices + (c + 4) * 9;
        if (!hiL) {  // K = 0,1
            A0.x = M0[0 + e]; A0.y = M0[3 + e];
            A1.x = M1[0 + e]; A1.y = M1[3 + e];
        } else {     // K = 2,3
            A0.x = M0[6 + e]; A0.y = biases[c * 3 + e];
            A1.x = M1[6 + e]; A1.y = biases[(c + 4) * 3 + e];
        }
    }

    float* wlds = lds + wib * (PTS_PER_WAVE * ROWSTRIDE);
    float* out_col = out + 3L * N;

    const long batch  = (long)WAVES_PER_BLOCK * PTS_PER_WAVE;        // 256 pts / block-iter
    const long stride = (long)gridDim.x * batch;
    long base = (long)blockIdx.x * batch + (long)wib * PTS_PER_WAVE;

    const int boff = hiL ? 1 : 0;   // lo lanes read (x,y); hi lanes read (y,z)

    for (; base + PTS_PER_WAVE <= N; base += stride) {
        // ---- B tiles: 4x16 f32, column = point, K = (x, y, z, 1). Branchless build:
        // one b64 load per half-batch per lane + cndmask selects (EXEC stays all-ones).
        const long nlo = base + m;
        f2 l0 = *(const f2*)(points + 3 * nlo + boff);
        f2 h0 = *(const f2*)(points + 3 * (nlo + 16) + boff);
        v2f Blo, Bhi;
        Blo.x = hiL ? l0.y : l0.x;   Blo.y = hiL ? 1.0f : l0.y;
        Bhi.x = hiL ? h0.y : h0.x;   Bhi.y = hiL ? 1.0f : h0.y;

        v8f z = {0.f, 0.f, 0.f, 0.f, 0.f, 0.f, 0.f, 0.f};
        // D[m][n] = component e of transform c applied to point n (+bias), all 8 transforms
        v8f Dll = __builtin_amdgcn_wmma_f32_16x16x4_f32(false, A0, false, Blo, (short)0, z, false, false);
        v8f Dhl = __builtin_amdgcn_wmma_f32_16x16x4_f32(false, A1, false, Blo, (short)0, z, false, false);
        v8f Dlh = __builtin_amdgcn_wmma_f32_16x16x4_f32(false, A0, false, Bhi, (short)0, z, false, false);
        v8f Dhh = __builtin_amdgcn_wmma_f32_16x16x4_f32(false, A1, false, Bhi, (short)0, z, false, false);

        // ---- Stage D into LDS as per-point columns of 32 rows (16B-aligned b128 stores)
        const int roff = hiL ? 8 : 0;  // C/D layout: lanes 16-31 hold rows v+8
        store8(wlds + (m     ) * ROWSTRIDE + roff,      Dll);  // rows 0..15  : transforms 0..3
        store8(wlds + (m     ) * ROWSTRIDE + 16 + roff, Dhl);  // rows 16..31 : transforms 4..7
        store8(wlds + (m + 16) * ROWSTRIDE + roff,      Dlh);
        store8(wlds + (m + 16) * ROWSTRIDE + 16 + roff, Dhh);

        // ---- Per-lane select + SELU + color mix. Three independent ds_loads,
        // single dscnt wait, branchless math.
        const long  p   = base + lane;
        const int   c   = choices[p];
        const int   rb  = ((c & 4) << 2) + 3 * (c & 3);  // row base of (c, e=0..2)
        const float* src = wlds + lane * ROWSTRIDE + rb;

        float s0 = src[0];
        float s1 = src[1];
        float s2 = src[2];

        const float* pc = prev_colors + 3 * p;
        float pc0 = pc[0], pc1 = pc[1], pc2 = pc[2];
        const float* cc = colors + 3 * c;
        float cc0 = cc[0], cc1 = cc[1], cc2 = cc[2];

        float* o = out + 3 * p;
        o[0] = selu_f(s0);
        o[1] = selu_f(s1);
        o[2] = selu_f(s2);

        float* oc = out_col + 3 * p;
        oc[0] = (pc0 + cc0) * 0.5f;
        oc[1] = (pc1 + cc1) * 0.5f;
        oc[2] = (pc2 + cc2) * 0.5f;
    }
}

extern "C" void kernel_launch(void* const* d_in, const int* in_sizes, int n_in,
                              void* d_out, int out_size, void* d_ws, size_t ws_size,
                              hipStream_t stream) {
    const float* points      = (const float*)d_in[0];
    const float* prev_colors = (const float*)d_in[1];
    const int*   choices     = (const int*)  d_in[2];
    const float* matrices    = (const float*)d_in[3];
    const float* biases      = (const float*)d_in[4];
    const float* colors      = (const float*)d_in[5];
    float* out = (float*)d_out;

    const int N = in_sizes[2];           // choices has N elements
    // Bandwidth-bound: modest grid + grid-stride loop amortizes A-tile build.
    int blocks = 4096;
    long need = ((long)N + 255) / 256;
    if ((long)blocks > need) blocks = (int)need;
    if (blocks < 1) blocks = 1;

    IFSFractal_wmma_kernel<<<blocks, 256, 0, stream>>>(
        points, prev_colors, choices, matrices, biases, colors, out, N);
}